// FCOS_11115375362152
// MI455X (gfx1250) — compile-verified
//
#include <hip/hip_runtime.h>
#include <hip/hip_bf16.h>
#include <math.h>
#include <stdint.h>

// ---------------- constants ----------------
#define NUM_CLASSES 80
#define CONF_THRESH 0.05f
#define NMS_THRESH  0.6f
#define CLASS_OFFSET 2.0f
#define NBINS 4096           // score histogram bins over [0,1)
#define CAP   4096           // per-level candidate capacity (bitonic size)
#define NTOT  3800           // 1000+1000+1000+640+160
#define NPAD  3808           // 238*16
#define NCC   238            // 16-wide column chunks covering NPAD

typedef float v2f __attribute__((ext_vector_type(2)));
typedef float v8f __attribute__((ext_vector_type(8)));

__device__ __forceinline__ float sigmoidf_(float x) {
    return 1.0f / (1.0f + __expf(-x));
}

// ---------------- K0: zero ----------------
__global__ void zero_kernel(unsigned* p, int n) {
    int t = blockIdx.x * blockDim.x + threadIdx.x;
    if (t < n) p[t] = 0u;
}

// ---------------- K1: fused score + histogram (one pass over cls tensor) ----------------
__global__ __launch_bounds__(256)
void hist_kernel(const float* __restrict__ cls, const float* __restrict__ ctn,
                 int hw, unsigned* __restrict__ hist) {
    __shared__ unsigned h[NBINS];
    for (int i = threadIdx.x; i < NBINS; i += blockDim.x) h[i] = 0u;
    __syncthreads();
    int total = hw * NUM_CLASSES;
    int stride = gridDim.x * blockDim.x;
    for (int e = blockIdx.x * blockDim.x + threadIdx.x; e < total; e += stride) {
        if (e + stride < total) __builtin_prefetch(&cls[e + stride], 0, 0);
        int a = e / NUM_CLASSES;
        float s = sqrtf(sigmoidf_(cls[e]) * sigmoidf_(ctn[a]));
        int b = (int)(s * (float)NBINS);
        b = b < 0 ? 0 : (b > NBINS - 1 ? NBINS - 1 : b);
        atomicAdd(&h[b], 1u);
    }
    __syncthreads();
    for (int i = threadIdx.x; i < NBINS; i += blockDim.x)
        if (h[i]) atomicAdd(&hist[i], h[i]);
}

// ---------------- K2: per-level threshold bin ----------------
__global__ void thresh_kernel(const unsigned* __restrict__ hist,
                              unsigned* __restrict__ thresh) {
    const int kk[5] = {1000, 1000, 1000, 640, 160};
    int L = blockIdx.x;
    if (threadIdx.x == 0) {
        const unsigned* h = hist + L * NBINS;
        unsigned acc = 0; int t = 0;
        for (int b = NBINS - 1; b >= 0; --b) {
            acc += h[b];
            if (acc >= (unsigned)kk[L]) { t = b; break; }
        }
        thresh[L] = (unsigned)t;
    }
}

// ---------------- K3: compact candidates above threshold ----------------
__global__ __launch_bounds__(256)
void compact_kernel(const float* __restrict__ cls, const float* __restrict__ ctn,
                    int hw, const unsigned* __restrict__ thresh, int L,
                    unsigned* __restrict__ cnt, float* __restrict__ cscore,
                    int* __restrict__ cidx) {
    unsigned th = thresh[L];
    int total = hw * NUM_CLASSES;
    int stride = gridDim.x * blockDim.x;
    for (int e = blockIdx.x * blockDim.x + threadIdx.x; e < total; e += stride) {
        if (e + stride < total) __builtin_prefetch(&cls[e + stride], 0, 0);
        int a = e / NUM_CLASSES;
        float s = sqrtf(sigmoidf_(cls[e]) * sigmoidf_(ctn[a]));
        int b = (int)(s * (float)NBINS);
        b = b < 0 ? 0 : (b > NBINS - 1 ? NBINS - 1 : b);
        if ((unsigned)b >= th) {
            unsigned p = atomicAdd(&cnt[L], 1u);
            if (p < CAP) {
                cscore[L * CAP + p] = s;
                cidx[L * CAP + p]   = e;
            }
        }
    }
}

// ---------------- LDS bitonic sort (descending by key), CAP elements ----------------
__device__ void bitonic_desc(float* key, int* val) {
    for (int k = 2; k <= CAP; k <<= 1) {
        for (int j = k >> 1; j > 0; j >>= 1) {
            for (int t = threadIdx.x; t < CAP; t += blockDim.x) {
                int ixj = t ^ j;
                if (ixj > t) {
                    bool up = ((t & k) == 0);   // descending overall
                    float a = key[t], b = key[ixj];
                    bool sw = up ? (a < b) : (a > b);
                    if (sw) {
                        key[t] = b; key[ixj] = a;
                        int v = val[t]; val[t] = val[ixj]; val[ixj] = v;
                    }
                }
            }
            __syncthreads();
        }
    }
}

// ---------------- K4: per-level sort + top-k emit ----------------
__global__ __launch_bounds__(1024)
void level_select_kernel(const float* __restrict__ box3, const float* __restrict__ box4,
                         const float* __restrict__ box5, const float* __restrict__ box6,
                         const float* __restrict__ box7,
                         const unsigned* __restrict__ cnt,
                         const float* __restrict__ cscore, const int* __restrict__ cidx,
                         float* __restrict__ scores_all, int* __restrict__ labels_all,
                         unsigned* __restrict__ valid_all,
                         float* __restrict__ boxes_all, float* __restrict__ boxesoff_all) {
    __shared__ float key[CAP];
    __shared__ int   val[CAP];
    const int kk[5] = {1000, 1000, 1000, 640, 160};
    const int oo[5] = {0, 1000, 2000, 3000, 3640};
    const int hh[5] = {40000, 10000, 2500, 640, 160};
    int L = blockIdx.x;
    const float* boxp = (L == 0) ? box3 : (L == 1) ? box4 : (L == 2) ? box5
                        : (L == 3) ? box6 : box7;
    int nc = (int)cnt[L]; if (nc > CAP) nc = CAP;
    for (int t = threadIdx.x; t < CAP; t += blockDim.x) {
        if (t < nc) { key[t] = cscore[L * CAP + t]; val[t] = cidx[L * CAP + t]; }
        else        { key[t] = -1e30f;              val[t] = -1; }
    }
    __syncthreads();
    bitonic_desc(key, val);
    int k = kk[L], o = oo[L], hwl = hh[L];
    for (int j = threadIdx.x; j < k; j += blockDim.x) {
        float s = key[j]; int idx = val[j];
        bool have = (idx >= 0);
        int label  = have ? (idx % NUM_CLASSES) : 0;
        int anchor = have ? (idx / NUM_CLASSES) : 0;
        if (anchor >= hwl) anchor = 0;
        bool valid = have && (s > CONF_THRESH);
        scores_all[o + j] = valid ? s : 0.0f;
        labels_all[o + j] = label;
        valid_all [o + j] = valid ? 1u : 0u;
        float offv = (float)label * CLASS_OFFSET;
        #pragma unroll
        for (int c = 0; c < 4; ++c) {
            float bx = boxp[anchor * 4 + c];
            boxes_all   [(o + j) * 4 + c] = bx;
            boxesoff_all[(o + j) * 4 + c] = bx + offv;
        }
    }
}

// ---------------- K5: global sort by score (argsort(-scores)) + gather ----------------
__global__ __launch_bounds__(1024)
void global_sort_kernel(const float* __restrict__ scores_all,
                        const unsigned* __restrict__ valid_all,
                        const float* __restrict__ boxesoff_all,
                        int* __restrict__ order, float* __restrict__ boxes_sorted,
                        unsigned* __restrict__ keepinit) {
    __shared__ float key[CAP];
    __shared__ int   val[CAP];
    for (int t = threadIdx.x; t < CAP; t += blockDim.x) {
        if (t < NTOT) { key[t] = scores_all[t]; val[t] = t; }
        else          { key[t] = -1e30f;        val[t] = -1; }
    }
    __syncthreads();
    bitonic_desc(key, val);
    for (int t = threadIdx.x; t < 240; t += blockDim.x) keepinit[t] = 0u;
    __syncthreads();
    for (int t = threadIdx.x; t < NPAD; t += blockDim.x) {
        int idx = (t < NTOT) ? val[t] : -1;
        order[t] = idx;
        #pragma unroll
        for (int c = 0; c < 4; ++c)
            boxes_sorted[t * 4 + c] = (idx >= 0) ? boxesoff_all[idx * 4 + c] : 0.0f;
        unsigned valid = (idx >= 0) ? valid_all[idx] : 0u;
        if (valid) atomicOr(&keepinit[t >> 4], 1u << (t & 15));
    }
}

// ---------------- K6: NMS pairwise tiles (wave32, async LDS staging + WMMA) --------
// One wave computes a 16x16 IoU tile. Row/col box tiles are staged
// global->LDS with global_load_async_to_lds_b128 (ASYNCcnt path).
// The union base (area_i + area_j) is produced on the matrix pipe with
// v_wmma_f32_16x16x4_f32: A(16x4)=[area_m,1,0,0], B(4x16)=[1;area_n;0;0].
__global__ __launch_bounds__(32)
void nms_tile_kernel(const float* __restrict__ boxes_sorted,
                     unsigned* __restrict__ supmat) {
    __shared__ float lds_boxes[128];   // [0:64) row boxes, [64:128) col boxes (16B each)
    int cb = blockIdx.x, rb = blockIdx.y;
    int lane = (int)threadIdx.x;
    int hi16 = lane >> 4;              // 0: stage row tile, 1: stage col tile
    int n    = lane & 15;              // column within tile

    // ---- async global->LDS staging of both 16-box tiles (256B each half) ----
    int src_row = hi16 ? (cb * 16 + n) : (rb * 16 + n);
    unsigned long long gaddr =
        (unsigned long long)(const void*)(boxes_sorted + src_row * 4);
    // Derive the LDS byte offset by ptrtoint of the real shared pointer:
    //  (a) generic LDS pointers carry the LDS byte offset in their low 32 bits
    //      (the LDS aperture uses the low 32 bits as the address), so this is the correct VDST;
    //  (b) ptrtoint CAPTURES lds_boxes, so the asm "memory" clobber forces all
    //      subsequent lds_boxes reads to stay after the async load + wait
    //      (without this, LLVM folds the reads to undef and hoists the WMMA).
    unsigned lds_off = (unsigned)(uintptr_t)(&lds_boxes[lane * 4]);
    asm volatile("global_load_async_to_lds_b128 %0, %1, off"
                 :: "v"(lds_off), "v"(gaddr) : "memory");
    asm volatile("s_wait_asynccnt 0x0" ::: "memory");
    __syncthreads();

    // col box for this lane's column n
    float4 cbox = *(const float4*)&lds_boxes[64 + n * 4];
    float carea = fmaxf(cbox.z - cbox.x, 0.0f) * fmaxf(cbox.w - cbox.y, 0.0f);
    // row area for A-matrix (lanes 0-15 carry M=lane)
    float4 rbx0 = *(const float4*)&lds_boxes[(lane & 15) * 4];
    float rarea = fmaxf(rbx0.z - rbx0.x, 0.0f) * fmaxf(rbx0.w - rbx0.y, 0.0f);

    // ---- WMMA outer-sum: U[m][n] = area_row_m + area_col_n ----
    // A 16x4 layout: vgpr0 = K0(lanes0-15)/K2(lanes16-31); vgpr1 = K1/K3
    // B 4x16 layout mirrors: vgpr0 = K0/K2 rows; vgpr1 = K1/K3 rows
    v2f A, B;
    A.x = hi16 ? 0.0f : rarea;  A.y = hi16 ? 0.0f : 1.0f;
    B.x = hi16 ? 0.0f : 1.0f;   B.y = hi16 ? 0.0f : carea;
    v8f C = {0.0f, 0.0f, 0.0f, 0.0f, 0.0f, 0.0f, 0.0f, 0.0f};
    v8f U = __builtin_amdgcn_wmma_f32_16x16x4_f32(
        /*neg_a=*/false, A, /*neg_b=*/false, B,
        /*c_mod=*/(short)0, C, /*reuse_a=*/false, /*reuse_b=*/false);

    int colg = cb * 16 + n;
    int rowbase = rb * 16;
    #pragma unroll
    for (int m = 0; m < 8; ++m) {
        int rlocal = m + hi16 * 8;     // D layout: vgpr m -> M=m (lanes0-15) / M=m+8
        float4 rbx = *(const float4*)&lds_boxes[rlocal * 4];
        float ix1 = fmaxf(rbx.x, cbox.x), iy1 = fmaxf(rbx.y, cbox.y);
        float ix2 = fminf(rbx.z, cbox.z), iy2 = fminf(rbx.w, cbox.w);
        float inter = fmaxf(ix2 - ix1, 0.0f) * fmaxf(iy2 - iy1, 0.0f);
        float uni = U[m] - inter;
        float iou = inter / fmaxf(uni, 1e-9f);
        int rowg = rowbase + rlocal;
        bool pred = (iou > NMS_THRESH) && (colg > rowg) &&
                    (colg < NTOT) && (rowg < NTOT);
        unsigned bal = __builtin_amdgcn_ballot_w32(pred);
        if (lane == 0)  supmat[(size_t)(rowbase + m) * NCC + cb]     = bal & 0xFFFFu;
        if (lane == 16) supmat[(size_t)(rowbase + m + 8) * NCC + cb] = bal >> 16;
    }
}

// ---------------- K7: sequential greedy NMS (keep bits in LDS) ----------------
__global__ __launch_bounds__(256)
void nms_seq_kernel(const unsigned* __restrict__ supmat,
                    const unsigned* __restrict__ keepinit,
                    unsigned* __restrict__ keep_sorted) {
    __shared__ unsigned keep[240];
    for (int t = threadIdx.x; t < 240; t += blockDim.x)
        keep[t] = (t < NCC) ? keepinit[t] : 0u;
    __syncthreads();
    for (int i = 0; i < NTOT; ++i) {
        bool on = (keep[i >> 4] >> (i & 15)) & 1u;
        __syncthreads();
        if (on) {
            const unsigned* row = supmat + (size_t)i * NCC;
            for (int t = threadIdx.x; t < NCC; t += blockDim.x)
                keep[t] &= ~row[t];
        }
        __syncthreads();
    }
    for (int t = threadIdx.x; t < NCC; t += blockDim.x)
        keep_sorted[t] = keep[t];
}

// ---------------- K8a: scatter keep through order ----------------
__global__ void scatter_keep_kernel(const int* __restrict__ order,
                                    const unsigned* __restrict__ keep_sorted,
                                    unsigned* __restrict__ keep_orig) {
    int t = blockIdx.x * blockDim.x + threadIdx.x;
    if (t >= NTOT) return;
    int o = order[t];
    if (o >= 0 && o < NTOT)
        keep_orig[o] = (keep_sorted[t >> 4] >> (t & 15)) & 1u;
}

// ---------------- K8b: final outputs (scores*keep, labels, boxes_n, keep) -------
__global__ void write_out_kernel(const float* __restrict__ scores_all,
                                 const int* __restrict__ labels_all,
                                 const float* __restrict__ boxes_all,
                                 const unsigned* __restrict__ keep_orig,
                                 const int* __restrict__ img_size,
                                 float* __restrict__ out) {
    int o = blockIdx.x * blockDim.x + threadIdx.x;
    if (o >= NTOT) return;
    float inv = 1.0f / (float)img_size[0];
    unsigned kp = keep_orig[o];
    out[o]        = scores_all[o] * (float)kp;
    out[NTOT + o] = (float)labels_all[o];
    #pragma unroll
    for (int c = 0; c < 4; ++c) {
        float v = boxes_all[o * 4 + c] * inv;
        v = fminf(fmaxf(v, 0.0f), 1.0f);
        out[2 * NTOT + o * 4 + c] = v;
    }
    out[6 * NTOT + o] = (float)kp;
}

// ---------------- host side ----------------
extern "C" void kernel_launch(void* const* d_in, const int* in_sizes, int n_in,
                              void* d_out, int out_size, void* d_ws, size_t ws_size,
                              hipStream_t stream) {
    (void)in_sizes; (void)n_in; (void)out_size; (void)ws_size;
    const int hws[5] = {40000, 10000, 2500, 640, 160};

    char* base = (char*)d_ws;
    size_t off = 0;
    auto alloc = [&](size_t bytes) -> char* {
        char* p = base + off;
        off = (off + bytes + 255) & ~(size_t)255;
        return p;
    };
    unsigned* hist        = (unsigned*)alloc(5 * NBINS * 4);
    unsigned* thresh      = (unsigned*)alloc(5 * 4);
    unsigned* cnt         = (unsigned*)alloc(5 * 4);
    float*    cscore      = (float*)   alloc(5 * CAP * 4);
    int*      cidx        = (int*)     alloc(5 * CAP * 4);
    float*    scores_all  = (float*)   alloc(NPAD * 4);
    int*      labels_all  = (int*)     alloc(NPAD * 4);
    unsigned* valid_all   = (unsigned*)alloc(NPAD * 4);
    float*    boxes_all   = (float*)   alloc(NPAD * 16);
    float*    boxesoff_all= (float*)   alloc(NPAD * 16);
    int*      order       = (int*)     alloc(NPAD * 4);
    float*    boxes_sorted= (float*)   alloc(NPAD * 16);
    unsigned* keepinit    = (unsigned*)alloc(240 * 4);
    unsigned* keep_sorted = (unsigned*)alloc(240 * 4);
    unsigned* keep_orig   = (unsigned*)alloc(NPAD * 4);
    unsigned* supmat      = (unsigned*)alloc((size_t)NPAD * NCC * 4);

    zero_kernel<<<(5 * NBINS + 255) / 256, 256, 0, stream>>>(hist, 5 * NBINS);
    zero_kernel<<<1, 256, 0, stream>>>(cnt, 5);

    for (int L = 0; L < 5; ++L) {
        const float* cls = (const float*)d_in[3 * L];
        const float* ctn = (const float*)d_in[3 * L + 1];
        int total = hws[L] * NUM_CLASSES;
        int blocks = (total + 255) / 256; if (blocks > 4096) blocks = 4096;
        hist_kernel<<<blocks, 256, 0, stream>>>(cls, ctn, hws[L], hist + L * NBINS);
    }
    thresh_kernel<<<5, 32, 0, stream>>>(hist, thresh);
    for (int L = 0; L < 5; ++L) {
        const float* cls = (const float*)d_in[3 * L];
        const float* ctn = (const float*)d_in[3 * L + 1];
        int total = hws[L] * NUM_CLASSES;
        int blocks = (total + 255) / 256; if (blocks > 4096) blocks = 4096;
        compact_kernel<<<blocks, 256, 0, stream>>>(cls, ctn, hws[L], thresh, L,
                                                   cnt, cscore, cidx);
    }
    level_select_kernel<<<5, 1024, 0, stream>>>(
        (const float*)d_in[2], (const float*)d_in[5], (const float*)d_in[8],
        (const float*)d_in[11], (const float*)d_in[14],
        cnt, cscore, cidx, scores_all, labels_all, valid_all,
        boxes_all, boxesoff_all);
    global_sort_kernel<<<1, 1024, 0, stream>>>(scores_all, valid_all, boxesoff_all,
                                               order, boxes_sorted, keepinit);
    nms_tile_kernel<<<dim3(NCC, NCC), 32, 0, stream>>>(boxes_sorted, supmat);
    nms_seq_kernel<<<1, 256, 0, stream>>>(supmat, keepinit, keep_sorted);
    scatter_keep_kernel<<<(NTOT + 255) / 256, 256, 0, stream>>>(order, keep_sorted,
                                                                keep_orig);
    write_out_kernel<<<(NTOT + 255) / 256, 256, 0, stream>>>(
        scores_all, labels_all, boxes_all, keep_orig,
        (const int*)d_in[15], (float*)d_out);
}